// GPool_1580547966979
// MI455X (gfx1250) — compile-verified
//
#include <hip/hip_runtime.h>
#include <math.h>

#define B   8
#define N   65536
#define DIM 128
#define K   16384

typedef float v2f __attribute__((ext_vector_type(2)));
typedef float v8f __attribute__((ext_vector_type(8)));

// ---------------------------------------------------------------------------
// Kernel 1: score y[b,n] = (x[b,n,:]·w + bias)/||w||  via V_WMMA_F32_16X16X4_F32
// One wave handles 16 rows. A-matrix layout (32-bit 16x4): lanes 0-15 hold
// K=0/1 in VGPR0/1, lanes 16-31 hold K=2/3. B is w broadcast across all 16
// columns (same half/K selector), so D[m][n] == raw score of row m for every
// n; lane 0 emits rows 0..7 (c[0..7]), lane 16 emits rows 8..15.
// Also builds 64-bit sort keys: ascending key == descending score, ties by
// ascending index (matches stable argsort(-y)).
// ---------------------------------------------------------------------------
__global__ void score_wmma(const float* __restrict__ x,
                           const float* __restrict__ w,
                           const float* __restrict__ bias,
                           float* __restrict__ y,
                           unsigned long long* __restrict__ keys) {
  const int lane  = threadIdx.x & 31;
  const int gwave = (int)((blockIdx.x * blockDim.x + threadIdx.x) >> 5);
  const int nwav  = (int)((gridDim.x * blockDim.x) >> 5);

  // 1/||w|| and bias (uniform; 128 scalar-cached loads)
  float nrm = 0.f;
#pragma unroll
  for (int i = 0; i < DIM; ++i) { float wv = w[i]; nrm += wv * wv; }
  const float inv_norm = 1.0f / sqrtf(nrm);
  const float bb = bias[0];

  const int half = lane >> 4;      // which half-wave
  const int m    = lane & 15;      // row within tile
  const int koff = half * 2;       // K offset inside group of 4

  const long long totalTiles = (long long)B * N / 16;
  for (long long tile = gwave; tile < totalTiles; tile += nwav) {
    const float* xr = x + (tile * 16 + m) * (long long)DIM;
    v8f c = {0.f, 0.f, 0.f, 0.f, 0.f, 0.f, 0.f, 0.f};
#pragma unroll
    for (int k0 = 0; k0 < DIM; k0 += 4) {
      v2f a, bv;
      a.x  = xr[k0 + koff];
      a.y  = xr[k0 + koff + 1];
      bv.x = w[k0 + koff];
      bv.y = w[k0 + koff + 1];
      c = __builtin_amdgcn_wmma_f32_16x16x4_f32(
              /*neg_a=*/false, a, /*neg_b=*/false, bv,
              /*c_mod=*/(short)0, c, /*reuse_a=*/false, /*reuse_b=*/false);
    }
    if (m == 0) {  // lane 0 -> rows 0..7, lane 16 -> rows 8..15
      const long long rowBase = tile * 16 + half * 8;
#pragma unroll
      for (int r = 0; r < 8; ++r) {
        const long long row = rowBase + r;        // global row in [0, B*N)
        const float yy = (c[r] + bb) * inv_norm;
        y[row] = yy;
        unsigned u = __float_as_uint(yy);
        u = (u & 0x80000000u) ? ~u : (u | 0x80000000u);  // monotone float->uint
        const unsigned kh = ~u;                           // descending order
        const unsigned nb = (unsigned)(row & (N - 1));    // index within batch
        keys[row] = ((unsigned long long)kh << 32) | nb;
      }
    }
  }
}

// ---------------------------------------------------------------------------
// Bitonic sort stages (ascending composite key per batch).
// Global stage for one (j,k) with j > 1024.
// ---------------------------------------------------------------------------
__global__ void bitonic_global(unsigned long long* __restrict__ keys,
                               int j, int k) {
  const long long t = blockIdx.x * (long long)blockDim.x + threadIdx.x; // pair id
  const long long i = (t / j) * (2LL * j) + (t % j);   // bit j of i is 0
  const long long p = i + j;
  const bool up = (((i & (N - 1)) & k) == 0);          // batch-local direction
  unsigned long long a = keys[i];
  unsigned long long c = keys[p];
  if (up ? (a > c) : (a < c)) { keys[i] = c; keys[p] = a; }
}

// All steps j = jstart .. 1 fused in LDS; block owns 2048 contiguous elements
// (never straddles a batch: N % 2048 == 0).
__global__ void bitonic_shared(unsigned long long* __restrict__ keys,
                               int jstart, int k) {
  __shared__ unsigned long long s[2048];
  const long long base = (long long)blockIdx.x * 2048;
  s[threadIdx.x]        = keys[base + threadIdx.x];
  s[threadIdx.x + 1024] = keys[base + threadIdx.x + 1024];
  __syncthreads();
  const int gbase = (int)(base & (N - 1));  // batch-local offset of this block
  for (int j = jstart; j >= 1; j >>= 1) {
    const int t = threadIdx.x;
    const int i = (t / j) * (2 * j) + (t % j);
    const int p = i + j;
    const bool up = (((gbase + i) & k) == 0);
    unsigned long long a = s[i];
    unsigned long long c = s[p];
    if (up ? (a > c) : (a < c)) { s[i] = c; s[p] = a; }
    __syncthreads();
  }
  keys[base + threadIdx.x]        = s[threadIdx.x];
  keys[base + threadIdx.x + 1024] = s[threadIdx.x + 1024];
}

// ---------------------------------------------------------------------------
// Kernel 3: emit outputs. One wave per output row (B*K rows).
// x_out row = x[b, idx, :] * sigmoid(y); 32 lanes x float4 = 128 floats.
// ---------------------------------------------------------------------------
__global__ void gather_out(const unsigned long long* __restrict__ keys,
                           const float* __restrict__ y,
                           const float* __restrict__ pos,
                           const float* __restrict__ x,
                           int* __restrict__ idx_out,
                           float* __restrict__ pos_out,
                           float* __restrict__ x_out) {
  const long long wave = (blockIdx.x * (long long)blockDim.x + threadIdx.x) >> 5;
  const int lane = threadIdx.x & 31;
  const long long b  = wave / K;
  const long long kk = wave % K;

  const unsigned long long key = keys[b * N + kk];
  const int idx = (int)(key & 0xFFFFFFFFu);
  const long long src = b * N + idx;

  const float yv = y[src];
  const float sc = 1.0f / (1.0f + expf(-yv));

  if (lane == 0) idx_out[wave] = idx;
  if (lane < 3)  pos_out[wave * 3 + lane] = pos[src * 3 + lane];

  const float4* xs = (const float4*)(x + src * DIM);
  float4* xd = (float4*)(x_out + wave * DIM);
  float4 v = xs[lane];
  v.x *= sc; v.y *= sc; v.z *= sc; v.w *= sc;
  xd[lane] = v;
}

// ---------------------------------------------------------------------------
extern "C" void kernel_launch(void* const* d_in, const int* in_sizes, int n_in,
                              void* d_out, int out_size, void* d_ws, size_t ws_size,
                              hipStream_t stream) {
  (void)in_sizes; (void)n_in; (void)out_size; (void)ws_size;
  const float* pos  = (const float*)d_in[0];  // [B,N,3]
  const float* x    = (const float*)d_in[1];  // [B,N,DIM]
  const float* w    = (const float*)d_in[2];  // [1,DIM]
  const float* bias = (const float*)d_in[3];  // [1]

  // workspace: keys (B*N u64, 4MB) | y (B*N f32, 2MB)
  unsigned long long* keys = (unsigned long long*)d_ws;
  float* y = (float*)((char*)d_ws + (size_t)B * N * sizeof(unsigned long long));

  // outputs, concatenated flat: top_idx[B,K] (int32 bits) | pos_out[B,K,3] | x_out[B,K,DIM]
  int*   idx_out = (int*)d_out;
  float* pos_out = (float*)d_out + (size_t)B * K;
  float* x_out   = pos_out + (size_t)B * K * 3;

  // 1) score + key build (WMMA GEMV), 16384 waves grid-striding 32768 tiles
  score_wmma<<<2048, 256, 0, stream>>>(x, w, bias, y, keys);

  // 2) per-batch ascending bitonic sort of composite keys
  for (int k = 2; k <= N; k <<= 1) {
    int j = k >> 1;
    for (; j > 1024; j >>= 1)
      bitonic_global<<<(B * N / 2) / 256, 256, 0, stream>>>(keys, j, k);
    bitonic_shared<<<(B * N) / 2048, 1024, 0, stream>>>(keys, j, k);
  }

  // 3) gather + scale: one wave per output row
  gather_out<<<(int)(((long long)B * K * 32) / 256), 256, 0, stream>>>(
      keys, y, pos, x, idx_out, pos_out, x_out);
}